// PointPillarScatter_8538394984457
// MI455X (gfx1250) — compile-verified
//
#include <hip/hip_runtime.h>
#include <stdint.h>
#include <stddef.h>

typedef float v4f __attribute__((ext_vector_type(4)));
typedef int   v4i_b128 __attribute__((vector_size(4 * sizeof(int))));

#ifndef __has_builtin
#define __has_builtin(x) 0
#endif

// CDNA5 async global->LDS path (ASYNCcnt). Guarded so the file compiles even
// if this toolchain spells the builtin differently; fallback uses plain loads.
#if defined(__AMDGCN__) && \
    __has_builtin(__builtin_amdgcn_global_load_async_to_lds_b128) && \
    __has_builtin(__builtin_amdgcn_s_wait_asynccnt)
#define PPS_ASYNC 1
#else
#define PPS_ASYNC 0
#endif

#define AS1 __attribute__((address_space(1)))
#define AS3 __attribute__((address_space(3)))

static constexpr int kNX    = 432;
static constexpr int kNY    = 496;
static constexpr int kPlane = kNX * kNY;   // 214272 cells per (b, c) plane
static constexpr int kCells = 1024;        // cells per block in gather kernel

// ---------------------------------------------------------------------------
// Gather path (preferred): map init -> index scatter -> coalesced gather
// ---------------------------------------------------------------------------

__global__ void pps_map_init(int* __restrict__ map, int M) {
    int i = blockIdx.x * blockDim.x + threadIdx.x;
    int stride = gridDim.x * blockDim.x;
    for (; i < M; i += stride) map[i] = -1;
}

__global__ void pps_map_scatter(const int* __restrict__ coords,
                                int* __restrict__ map, int N) {
    int p = blockIdx.x * blockDim.x + threadIdx.x;
    if (p >= N) return;
    int b = coords[p * 4 + 0];
    int z = coords[p * 4 + 1];
    int y = coords[p * 4 + 2];
    int x = coords[p * 4 + 3];
    map[b * kPlane + z + y * kNX + x] = p;
}

// Each block: one batch b, kCells consecutive cells, ALL C channels.
// Map tile staged in LDS (async on CDNA5); output written once, coalesced,
// with non-temporal b128 stores (output >> L2, never re-read).
__global__ void pps_gather(const float* __restrict__ feats,
                           const int* __restrict__ map,
                           float* __restrict__ out, int C) {
    __shared__ int smap[kCells];
    const int b        = blockIdx.y;
    const int cellBase = blockIdx.x * kCells;
    const int t        = threadIdx.x;          // 0..255, 4 cells per thread
    const int* gmap    = map + (size_t)b * kPlane + cellBase;
    const int c0       = 4 * t;

#if PPS_ASYNC
    if (cellBase + c0 + 4 <= kPlane) {
        __builtin_amdgcn_global_load_async_to_lds_b128(
            (AS1 v4i_b128*)((AS1 const int*)gmap + c0),
            (AS3 v4i_b128*)((AS3 int*)smap + c0), 0, 0);
    } else {
        #pragma unroll
        for (int j = 0; j < 4; ++j) {
            int cc = c0 + j;
            smap[cc] = (cellBase + cc < kPlane) ? gmap[cc] : -1;
        }
    }
    __builtin_amdgcn_s_wait_asynccnt(0);
#else
    #pragma unroll
    for (int j = 0; j < 4; ++j) {
        int cc = c0 + j;
        smap[cc] = (cellBase + cc < kPlane) ? gmap[cc] : -1;
    }
#endif
    __syncthreads();

    const int cell0 = cellBase + c0;
    float* obase = out + ((size_t)b * C) * kPlane;
    const bool full = (cell0 + 4 <= kPlane);

    for (int c = 0; c < C; ++c) {
        float* orow = obase + (size_t)c * kPlane;
        if (full) {
            v4f v;
            #pragma unroll
            for (int j = 0; j < 4; ++j) {
                int idx = smap[c0 + j];  // feats resident in 192MB L2 -> cheap gather
                v[j] = (idx >= 0) ? feats[(size_t)idx * C + c] : 0.0f;
            }
            __builtin_nontemporal_store(v, (v4f*)(orow + cell0));
        } else {
            for (int j = 0; j < 4; ++j) {
                int cell = cell0 + j;
                if (cell < kPlane) {
                    int idx = smap[c0 + j];
                    float v = (idx >= 0) ? feats[(size_t)idx * C + c] : 0.0f;
                    __builtin_nontemporal_store(v, orow + cell);
                }
            }
        }
    }
}

// ---------------------------------------------------------------------------
// Fallback path (if d_ws can't hold the map): zero-fill + direct scatter
// ---------------------------------------------------------------------------

__global__ void pps_zero(v4f* __restrict__ out4, long long nvec, long long n) {
    long long i = (long long)blockIdx.x * blockDim.x + threadIdx.x;
    long long stride = (long long)gridDim.x * blockDim.x;
    v4f z = {0.0f, 0.0f, 0.0f, 0.0f};
    for (long long v = i; v < nvec; v += stride)
        __builtin_nontemporal_store(z, out4 + v);
    for (long long s = nvec * 4 + i; s < n; s += stride)
        ((float*)out4)[s] = 0.0f;
}

// One pillar per wave32: lanes cover channels, 256B coalesced feature read,
// strided channel-plane stores.
__global__ void pps_scatter(const float* __restrict__ feats,
                            const int* __restrict__ coords,
                            float* __restrict__ out, int N, int C) {
    int pillar = blockIdx.x * (blockDim.x >> 5) + (threadIdx.x >> 5);
    if (pillar >= N) return;
    int lane = threadIdx.x & 31;
    int b = coords[pillar * 4 + 0];
    int z = coords[pillar * 4 + 1];
    int y = coords[pillar * 4 + 2];
    int x = coords[pillar * 4 + 3];
    long long base = ((long long)b * C) * kPlane + z + (long long)y * kNX + x;
    for (int c = lane; c < C; c += 32) {
        float v = feats[(size_t)pillar * C + c];
        __builtin_nontemporal_store(v, out + base + (long long)c * kPlane);
    }
}

// ---------------------------------------------------------------------------

extern "C" void kernel_launch(void* const* d_in, const int* in_sizes, int n_in,
                              void* d_out, int out_size, void* d_ws, size_t ws_size,
                              hipStream_t stream) {
    const float* feats  = (const float*)d_in[0];
    const int*   coords = (const int*)d_in[1];
    float*       out    = (float*)d_out;

    const int N = in_sizes[1] / 4;                 // pillars
    const int C = (N > 0) ? (in_sizes[0] / N) : 64;  // channels (64)
    const long long M_ll = (C > 0) ? ((long long)out_size / C) : 0;  // B*plane
    const int M = (int)M_ll;
    const size_t mapBytes = (size_t)M * sizeof(int);

    const bool gatherOK = (M > 0) && ((long long)M * C == (long long)out_size) &&
                          (M % kPlane == 0) && (ws_size >= mapBytes);

    if (gatherOK) {
        int* map = (int*)d_ws;
        const int B = M / kPlane;
        {
            int threads = 256;
            int blocks = (M + threads - 1) / threads;
            pps_map_init<<<blocks, threads, 0, stream>>>(map, M);
        }
        if (N > 0)
            pps_map_scatter<<<(N + 255) / 256, 256, 0, stream>>>(coords, map, N);
        dim3 grid((kPlane + kCells - 1) / kCells, B);
        pps_gather<<<grid, 256, 0, stream>>>(feats, map, out, C);
    } else {
        long long n = out_size;
        long long nvec = n / 4;
        pps_zero<<<4096, 256, 0, stream>>>((v4f*)out, nvec, n);
        if (N > 0) {
            int blocks = (N + 7) / 8;   // 8 waves per 256-thread block
            pps_scatter<<<blocks, 256, 0, stream>>>(feats, coords, out, N, C);
        }
    }
}